// escn_tensor_product_42090679501093
// MI455X (gfx1250) — compile-verified
//
#include <hip/hip_runtime.h>

typedef __attribute__((ext_vector_type(2))) float v2f;
typedef __attribute__((ext_vector_type(8))) float v8f;

#define XB 49
#define OB 49
#define CCH 128
#define NPAIR 18
#define WSTRIDE 66          // padded row stride of W in LDS (floats); 66 mod 64 = 2 -> conflict-free A loads
#define AENTRIES 1170       // sum over pairs of (2*out_l+1)*(2*l+1)

// ---- compile-time entry -> pair lookup (replaces runtime linear search) ----
struct PairTab {
    unsigned char  l[NPAIR];      // l of pair k
    unsigned char  ol[NPAIR];     // out_l of pair k
    unsigned short base[NPAIR];   // flat-entry offset of pair k
    unsigned char  ek[AENTRIES];  // flat entry -> pair index k
    constexpr PairTab() : l{}, ol{}, base{}, ek{} {
        const int L[NPAIR] = {0, 1,1,1, 2,2,2, 3,3,3, 4,4,4, 5,5,5, 6,6};
        const int O[NPAIR] = {1, 0,1,2, 1,2,3, 2,3,4, 3,4,5, 4,5,6, 5,6};
        int off = 0;
        for (int k = 0; k < NPAIR; ++k) {
            l[k]  = (unsigned char)L[k];
            ol[k] = (unsigned char)O[k];
            base[k] = (unsigned short)off;
            const int sz = (2 * O[k] + 1) * (2 * L[k] + 1);
            for (int e = 0; e < sz; ++e) ek[off + e] = (unsigned char)k;
            off += sz;
        }
    }
};
__constant__ PairTab PT = PairTab();

__global__ __launch_bounds__(256)
void escn_tp_kernel(const float* __restrict__ t1,
                    const float* __restrict__ t2,
                    const float* __restrict__ R,
                    const float* __restrict__ wD,
                    const float* __restrict__ CG,
                    const float* __restrict__ wts,
                    float* __restrict__ out,
                    int N)
{
    __shared__ float sD[XB * XB];        // wigner_D[n], row-major 49x49
    __shared__ float sCG[XB * OB];       // CG[:,1,:]   49x49
    __shared__ float sA[AENTRIES];       // per-pair A = right^T @ D_l
    __shared__ float sW[64 * WSTRIDE];   // fused 49x49 matrix, padded to 64x66
    __shared__ float sWk[NPAIR];         // w_k = t2m * weights[k] * 0.5

    const int n   = blockIdx.x;
    const int tid = threadIdx.x;
    if (n >= N) return;

    // ---------- build phase: W[n] = sum_k w_k * P_out (D_out^T right^T D_l) P_l^T ----------
    for (int i = tid; i < 64 * WSTRIDE; i += 256) sW[i] = 0.0f;
    for (int i = tid; i < XB * XB; i += 256) sD[i] = wD[(size_t)n * (XB * XB) + i];
    for (int i = tid; i < XB * OB; i += 256) {
        int j = i / OB, o = i % OB;                 // CG[j, 1, o]
        sCG[i] = CG[(size_t)j * (3 * OB) + OB + o];
    }
    if (tid < NPAIR) {
        float t2m = 0.0f;                           // (R[n] @ t2[n])[1]
        for (int j = 0; j < 3; ++j) t2m += R[(size_t)n * 9 + 3 + j] * t2[(size_t)n * 3 + j];
        sWk[tid] = t2m * wts[tid] * 0.5f;
    }
    __syncthreads();

    // pass 1: A_k[o][j] = sum_q CG[s+q][os+o] * D[s+q][s+j]
    for (int e = tid; e < AENTRIES; e += 256) {
        const int k     = PT.ek[e];
        const int l     = PT.l[k];
        const int ol    = PT.ol[k];
        const int base  = PT.base[k];
        const int ncols = 2 * l + 1;
        const int loc   = e - base;
        const int o = loc / ncols, j = loc % ncols;
        const int s = l * l, os = ol * ol;
        float acc = 0.0f;
        for (int q = 0; q < ncols; ++q)
            acc += sCG[(s + q) * OB + (os + o)] * sD[(s + q) * XB + (s + j)];
        sA[e] = acc;
    }
    __syncthreads();

    // pass 2: W[os+i][s+j] = w_k * sum_p D[os+p][os+i] * A_k[p][j]
    for (int e = tid; e < AENTRIES; e += 256) {
        const int k     = PT.ek[e];
        const int l     = PT.l[k];
        const int ol    = PT.ol[k];
        const int base  = PT.base[k];
        const int ncols = 2 * l + 1;
        const int nrows = 2 * ol + 1;
        const int loc   = e - base;
        const int i = loc / ncols, j = loc % ncols;
        const int s = l * l, os = ol * ol;
        float acc = 0.0f;
        for (int p = 0; p < nrows; ++p)
            acc += sD[(os + p) * XB + (os + i)] * sA[base + p * ncols + j];
        sW[(os + i) * WSTRIDE + (s + j)] = sWk[k] * acc;
    }
    __syncthreads();

    // ---------- GEMM phase: res[n] = W @ t1[n]  (49x49 x 49x128, fp32 WMMA) ----------
    const int lane = tid & 31;
    const int wave = tid >> 5;               // 8 waves = 8 column tiles of 16
    const int rsel = lane >> 4;              // 0 or 1 (half-wave)
    const int m16  = lane & 15;
    const int col  = wave * 16 + m16;
    const float* t1n = t1 + (size_t)n * XB * CCH;

    // block-tridiagonal zero-tile mask: nz[mtile][ktile]
    constexpr bool nz[4][4] = {
        {true,  true,  false, false},
        {true,  true,  true,  true },
        {false, true,  true,  true },
        {false, false, true,  true }
    };

    v8f acc[4];
    #pragma unroll
    for (int m = 0; m < 4; ++m) acc[m] = (v8f){0.f,0.f,0.f,0.f,0.f,0.f,0.f,0.f};

    auto mma_tile = [&](v8f c, int mt, int kt, const v2f* b) -> v8f {
        #pragma unroll
        for (int ks = 0; ks < 4; ++ks) {
            const int kk = kt * 16 + ks * 4 + 2 * rsel;     // K index of .x element
            const int ai = (mt * 16 + m16) * WSTRIDE + kk;  // A: W[m16][kk], W[m16][kk+1]
            v2f a;
            a.x = sW[ai];
            a.y = sW[ai + 1];
            c = __builtin_amdgcn_wmma_f32_16x16x4_f32(
                    /*neg_a=*/false, a, /*neg_b=*/false, b[ks],
                    /*c_mod=*/(short)0, c, /*reuse_a=*/false, /*reuse_b=*/false);
        }
        return c;
    };

    #pragma unroll
    for (int kt = 0; kt < 4; ++kt) {
        // load B fragments for this K tile (t1 rows kt*16 .. kt*16+15), rows >= 49 -> 0
        v2f b[4];
        #pragma unroll
        for (int ks = 0; ks < 4; ++ks) {
            const int r0 = kt * 16 + ks * 4 + 2 * rsel;
            float b0 = 0.0f, b1 = 0.0f;
            if (r0 < XB)     b0 = t1n[(size_t)r0 * CCH + col];
            if (r0 + 1 < XB) b1 = t1n[(size_t)(r0 + 1) * CCH + col];
            v2f bb; bb.x = b0; bb.y = b1;
            b[ks] = bb;
        }
        #pragma unroll
        for (int mt = 0; mt < 4; ++mt) {
            if (nz[mt][kt]) acc[mt] = mma_tile(acc[mt], mt, kt, b);
        }
    }

    // store: D fragment element v -> row = mt*16 + v + 8*rsel, col as above
    float* outn = out + (size_t)n * OB * CCH;
    #pragma unroll
    for (int mt = 0; mt < 4; ++mt) {
        #pragma unroll
        for (int v = 0; v < 8; ++v) {
            const int row = mt * 16 + v + 8 * rsel;
            if (row < OB) outn[(size_t)row * CCH + col] = acc[mt][v];
        }
    }
}

extern "C" void kernel_launch(void* const* d_in, const int* in_sizes, int n_in,
                              void* d_out, int out_size, void* d_ws, size_t ws_size,
                              hipStream_t stream)
{
    (void)n_in; (void)d_ws; (void)ws_size; (void)out_size;
    const float* t1  = (const float*)d_in[0];
    const float* t2  = (const float*)d_in[1];
    const float* R   = (const float*)d_in[2];
    const float* wD  = (const float*)d_in[3];
    const float* CG  = (const float*)d_in[4];
    const float* wts = (const float*)d_in[5];
    float* out = (float*)d_out;

    const int N = in_sizes[0] / (XB * CCH);
    escn_tp_kernel<<<dim3(N), dim3(256), 0, stream>>>(t1, t2, R, wD, CG, wts, out, N);
}